// Attention_69733089018373
// MI455X (gfx1250) — compile-verified
//
#include <hip/hip_runtime.h>

// Bilinear attention (no softmax) reformulated:
//   out_b = x_b * M_b,  M_b = sum_h Wq_h (Wk_h^T (x_b^T x_b) Wv_h) Wo_h
// f32 GEMMs on V_WMMA_F32_16X16X4_F32. Heavy GEMMs (x^T x and x*M) use the
// Tensor Data Mover to stage padded K-panels into LDS (double buffered,
// TENSORcnt-synchronized); 8 waves per workgroup share the panels.

typedef __attribute__((ext_vector_type(2))) float v2f;
typedef __attribute__((ext_vector_type(8))) float v8f;
typedef __attribute__((ext_vector_type(4))) unsigned int v4u;
typedef __attribute__((ext_vector_type(8))) int v8i;
typedef __attribute__((ext_vector_type(4))) int v4i;

#define B_    2
#define L_    4096
#define DIN   512
#define H_    8
#define DK    64
#define DV    64
#define DOUT  512

__device__ __forceinline__ v8f wmma4(v2f a, v2f b, v8f c) {
  return __builtin_amdgcn_wmma_f32_16x16x4_f32(false, a, false, b, (short)0, c,
                                               false, false);
}

// ---------------------------------------------------------------------------
// TDM: issue a 2D tensor_load_to_lds. D# built per CDNA5 ISA §8.3/8.4.
//   d0 = contiguous elements per row, d1 = rows, stride0 = row stride (elems),
//   pad_int_code: LDS pad interval (0=2,1=4,...,5=64,6=128 DWORDs),
//   pad_amt_code: pad amount minus 1 (3 => 4 DWORDs).
// f32 elements (data_size code 2). clang-23 builtin form: 6 args
//   (v4u group0, v8i group1, v4i group2, v4i group3, v8i group4, i32 cpol).
// ---------------------------------------------------------------------------
__device__ __forceinline__ void tdm_load_2d(unsigned lds_addr, const void* gptr,
                                            unsigned d0, unsigned d1,
                                            unsigned stride0,
                                            unsigned pad_int_code,
                                            unsigned pad_amt_code) {
  unsigned long long ga = (unsigned long long)(size_t)gptr;
  v4u g0;
  g0[0] = 1u;                                          // count=1, user D#
  g0[1] = lds_addr;                                    // LDS byte address
  g0[2] = (unsigned)(ga & 0xFFFFFFFFu);                // global_addr[31:0]
  g0[3] = (unsigned)((ga >> 32) & 0x1FFFFFFu)          // global_addr[56:32]
        | (2u << 30);                                  // type = 2 ("image")
  unsigned w0 = (2u << 16)                             // data_size = 4 bytes
              | (1u << 20)                             // pad_enable
              | (pad_int_code << 22)
              | (pad_amt_code << 25);
  unsigned long long s0 = stride0;
  v8i g1;
  g1[0] = (int)w0;                                     // mask=0, flags
  g1[1] = (int)((d0 & 0xFFFFu) << 16);                 // tensor_dim0[15:0]
  g1[2] = (int)((d0 >> 16) | ((d1 & 0xFFFFu) << 16));  // tdim0 hi | tdim1 lo
  g1[3] = (int)((d1 >> 16) | (d0 << 16));              // tdim1 hi | tile_dim0
  g1[4] = (int)(d1 & 0xFFFFu);                         // tile_dim1 (tile_dim2=0)
  g1[5] = (int)(s0 & 0xFFFFFFFFu);                     // tensor_dim0_stride lo
  g1[6] = (int)((s0 >> 32) & 0xFFFFu);                 // stride0 hi | stride1 lo=0
  g1[7] = 0;                                           // stride1 hi = 0
  v4i z4 = {0, 0, 0, 0};                               // groups 2/3: unused dims
  v8i z8 = {0, 0, 0, 0, 0, 0, 0, 0};                   // group 4: unused
  __builtin_amdgcn_tensor_load_to_lds(g0, g1, z4, z4, z8, 0);
}

// ---------------------------------------------------------------------------
// Direct-global 16x64 strip (for the small mid-chain GEMMs).
// ---------------------------------------------------------------------------
template <bool TRANS_A>
__device__ __forceinline__ void gemm_strip(const float* __restrict__ A,
                                           const float* __restrict__ Bm,
                                           float* __restrict__ C,
                                           int m0, int n0, int K,
                                           int lda, int ldb, int ldc) {
  const int lane = threadIdx.x & 31;
  const int half = lane >> 4, mn = lane & 15;
  v8f acc0 = {}, acc1 = {}, acc2 = {}, acc3 = {};
  for (int k = 0; k < K; k += 4) {
    v2f a;
    if (TRANS_A) {
      const float* ap = A + (size_t)(k + 2 * half) * lda + (m0 + mn);
      a.x = ap[0]; a.y = ap[lda];
    } else {
      const float* ap = A + (size_t)(m0 + mn) * lda + (k + 2 * half);
      a.x = ap[0]; a.y = ap[1];
    }
    const float* bp = Bm + (size_t)(k + 2 * half) * ldb + (n0 + mn);
    v2f b0, b1, b2, b3;
    b0.x = bp[0];  b0.y = bp[ldb];
    b1.x = bp[16]; b1.y = bp[ldb + 16];
    b2.x = bp[32]; b2.y = bp[ldb + 32];
    b3.x = bp[48]; b3.y = bp[ldb + 48];
    acc0 = wmma4(a, b0, acc0);
    acc1 = wmma4(a, b1, acc1);
    acc2 = wmma4(a, b2, acc2);
    acc3 = wmma4(a, b3, acc3);
  }
  float* cp = C + (size_t)(m0 + half * 8) * ldc + (n0 + mn);
#pragma unroll
  for (int r = 0; r < 8; ++r) {
    cp[0] = acc0[r]; cp[16] = acc1[r]; cp[32] = acc2[r]; cp[48] = acc3[r];
    cp += ldc;
  }
}

// Shared 4-WMMA inner step reading fragments from padded LDS panels.
__device__ __forceinline__ void lds_step(const float* ap, int a_stride,
                                         const float* bp, int ldsb,
                                         v8f& a0, v8f& a1, v8f& a2, v8f& a3) {
  v2f a, b0, b1, b2, b3;
  a.x = ap[0]; a.y = ap[a_stride];
  b0.x = bp[0];  b0.y = bp[ldsb];
  b1.x = bp[16]; b1.y = bp[ldsb + 16];
  b2.x = bp[32]; b2.y = bp[ldsb + 32];
  b3.x = bp[48]; b3.y = bp[ldsb + 48];
  a0 = wmma4(a, b0, a0); a1 = wmma4(a, b1, a1);
  a2 = wmma4(a, b2, a2); a3 = wmma4(a, b3, a3);
}

// ---------------------------------------------------------------------------
// S[b] = x_b^T x_b  (512x512, K=4096). Block: 128x64 tile, 8 waves.
// LDS panels: lA = x[k0:k0+64][i0:i0+128] (row stride 132 = 128+4 pad)
//             lB = x[k0:k0+64][j0:j0+64]  (row stride 68  = 64+4 pad)
// grid (8, 4, 2) x 256 threads
// ---------------------------------------------------------------------------
#define XT_LDA 132
#define XT_LDB 68
__global__ void __launch_bounds__(256) k_xtx(const float* __restrict__ x,
                                             float* __restrict__ S) {
  __shared__ float lA[2][64 * XT_LDA];
  __shared__ float lB[2][64 * XT_LDB];
  const int b = blockIdx.z;
  const int i0 = blockIdx.y * 128, j0 = blockIdx.x * 64;
  const float* xb = x + (size_t)b * L_ * DIN;
  const int wave = threadIdx.x >> 5, lane = threadIdx.x & 31;
  const int half = lane >> 4, mn = lane & 15;
  const int m0 = wave * 16;

  auto issue = [&](int buf, int k0) {
    tdm_load_2d((unsigned)(size_t)&lA[buf][0], xb + (size_t)k0 * DIN + i0,
                128, 64, DIN, /*pad every 128 DW*/ 6, /*4 DW*/ 3);
    tdm_load_2d((unsigned)(size_t)&lB[buf][0], xb + (size_t)k0 * DIN + j0,
                64, 64, DIN, /*pad every 64 DW*/ 5, /*4 DW*/ 3);
  };

  if (wave == 0) {
    issue(0, 0);
    __builtin_amdgcn_s_wait_tensorcnt(0);
  }
  __syncthreads();

  v8f acc0 = {}, acc1 = {}, acc2 = {}, acc3 = {};
  const int nch = L_ / 64;
  for (int c = 0; c < nch; ++c) {
    const float* A = &lA[c & 1][0];
    const float* Bm = &lB[c & 1][0];
    const bool more = (c + 1) < nch;
    if (wave == 0 && more) issue((c + 1) & 1, (c + 1) * 64);  // overlap DMA
#pragma unroll 4
    for (int kk = 0; kk < 64; kk += 4) {
      // A operand is x^T: A_frag[m][k] = x[k][i0+m] -> k-major LDS panel
      lds_step(A + (kk + 2 * half) * XT_LDA + m0 + mn, XT_LDA,
               Bm + (kk + 2 * half) * XT_LDB + mn, XT_LDB,
               acc0, acc1, acc2, acc3);
    }
    if (wave == 0 && more) __builtin_amdgcn_s_wait_tensorcnt(0);
    __syncthreads();
  }

  float* cp = S + (size_t)b * DIN * DIN + (size_t)(i0 + m0 + half * 8) * DIN
            + j0 + mn;
#pragma unroll
  for (int r = 0; r < 8; ++r) {
    cp[0] = acc0[r]; cp[16] = acc1[r]; cp[32] = acc2[r]; cp[48] = acc3[r];
    cp += DIN;
  }
}

// ---------------------------------------------------------------------------
// out[b] = x_b * M_b  (4096x512, K=512). Block: 128x64 tile, 8 waves.
// LDS panels: lA = x[r0:r0+128][k0:k0+64] (m-major, row stride 68)
//             lB = M[k0:k0+64][n0:n0+64]  (k-major, row stride 68)
// grid (8, 32, 2) x 256 threads
// ---------------------------------------------------------------------------
#define XO_LD 68
__global__ void __launch_bounds__(256) k_out(const float* __restrict__ x,
                                             const float* __restrict__ M,
                                             float* __restrict__ out) {
  __shared__ float lA[2][128 * XO_LD];
  __shared__ float lB[2][64 * XO_LD];
  const int b = blockIdx.z;
  const int r0 = blockIdx.y * 128, n0 = blockIdx.x * 64;
  const float* xb = x + (size_t)b * L_ * DIN;
  const float* Mb = M + (size_t)b * DIN * DOUT;
  const int wave = threadIdx.x >> 5, lane = threadIdx.x & 31;
  const int half = lane >> 4, mn = lane & 15;
  const int m0 = wave * 16;

  auto issue = [&](int buf, int k0) {
    tdm_load_2d((unsigned)(size_t)&lA[buf][0], xb + (size_t)r0 * DIN + k0,
                64, 128, DIN, 5, 3);
    tdm_load_2d((unsigned)(size_t)&lB[buf][0], Mb + (size_t)k0 * DOUT + n0,
                64, 64, DOUT, 5, 3);
  };

  if (wave == 0) {
    issue(0, 0);
    __builtin_amdgcn_s_wait_tensorcnt(0);
  }
  __syncthreads();

  v8f acc0 = {}, acc1 = {}, acc2 = {}, acc3 = {};
  const int nch = DIN / 64;
  for (int c = 0; c < nch; ++c) {
    const float* A = &lA[c & 1][0];
    const float* Bm = &lB[c & 1][0];
    const bool more = (c + 1) < nch;
    if (wave == 0 && more) issue((c + 1) & 1, (c + 1) * 64);
#pragma unroll 4
    for (int kk = 0; kk < 64; kk += 4) {
      // A_frag[m][k] = x[r0+m][k] -> m-major panel, contiguous k pair
      lds_step(A + (m0 + mn) * XO_LD + kk + 2 * half, 1,
               Bm + (kk + 2 * half) * XO_LD + mn, XO_LD,
               acc0, acc1, acc2, acc3);
    }
    if (wave == 0 && more) __builtin_amdgcn_s_wait_tensorcnt(0);
    __syncthreads();
  }

  float* cp = out + (size_t)b * L_ * DOUT + (size_t)(r0 + m0 + half * 8) * DOUT
            + n0 + mn;
#pragma unroll
  for (int r = 0; r < 8; ++r) {
    cp[0] = acc0[r]; cp[16] = acc1[r]; cp[32] = acc2[r]; cp[48] = acc3[r];
    cp += DOUT;
  }
}

// --------------------- small mid-chain kernels (direct) ---------------------
// T1[b,h] = Wk_h^T S_b  (64x512, K=512)          grid (8, 4, 16)
__global__ void k_t1(const float* __restrict__ Wk, const float* __restrict__ S,
                     float* __restrict__ T1) {
  const int bh = blockIdx.z, b = bh >> 3, h = bh & 7;
  gemm_strip<true>(Wk + (size_t)h * DIN * DK, S + (size_t)b * DIN * DIN,
                   T1 + (size_t)bh * DK * DIN,
                   blockIdx.y * 16, blockIdx.x * 64, DIN, DK, DIN, DIN);
}

// KTV[b,h] = T1[b,h] Wv_h  (64x64, K=512)        grid (1, 4, 16)
__global__ void k_ktv(const float* __restrict__ T1, const float* __restrict__ Wv,
                      float* __restrict__ KTV) {
  const int bh = blockIdx.z, h = bh & 7;
  gemm_strip<false>(T1 + (size_t)bh * DK * DIN, Wv + (size_t)h * DIN * DV,
                    KTV + (size_t)bh * DK * DV,
                    blockIdx.y * 16, 0, DIN, DIN, DV, DV);
}

// P2[b][:, h*64:(h+1)*64] = Wq_h KTV[b,h]  (512x64, K=64)   grid (1, 32, 16)
__global__ void k_p2(const float* __restrict__ Wq, const float* __restrict__ KTV,
                     float* __restrict__ P2) {
  const int bh = blockIdx.z, b = bh >> 3, h = bh & 7;
  gemm_strip<false>(Wq + (size_t)h * DIN * DK, KTV + (size_t)bh * DK * DV,
                    P2 + (size_t)b * DIN * (H_ * DV) + h * DV,
                    blockIdx.y * 16, 0, DK, DK, DV, H_ * DV);
}

// M[b] = P2[b] * WoFlat  (512x512, K=512)        grid (8, 32, 2)
__global__ void k_m(const float* __restrict__ P2, const float* __restrict__ Wo,
                    float* __restrict__ M) {
  const int b = blockIdx.z;
  gemm_strip<false>(P2 + (size_t)b * DIN * DIN, Wo, M + (size_t)b * DIN * DIN,
                    blockIdx.y * 16, blockIdx.x * 64, H_ * DV, H_ * DV, DOUT, DOUT);
}

extern "C" void kernel_launch(void* const* d_in, const int* in_sizes, int n_in,
                              void* d_out, int out_size, void* d_ws, size_t ws_size,
                              hipStream_t stream) {
  const float* x  = (const float*)d_in[0];
  const float* Wq = (const float*)d_in[1];
  const float* Wk = (const float*)d_in[2];
  const float* Wv = (const float*)d_in[3];
  const float* Wo = (const float*)d_in[4];
  float* out = (float*)d_out;

  // Workspace layout (floats): ~8.7 MB total
  float* S   = (float*)d_ws;                     // 2*512*512
  float* T1  = S   + 2 * 512 * 512;              // 2*8*64*512
  float* KTV = T1  + 2 * 8 * 64 * 512;           // 2*8*64*64
  float* P2  = KTV + 2 * 8 * 64 * 64;            // 2*512*512
  float* M   = P2  + 2 * 512 * 512;              // 2*512*512

  k_xtx<<<dim3(8, 4, 2),   dim3(256), 0, stream>>>(x, S);
  k_t1 <<<dim3(8, 4, 16),  dim3(32),  0, stream>>>(Wk, S, T1);
  k_ktv<<<dim3(1, 4, 16),  dim3(32),  0, stream>>>(T1, Wv, KTV);
  k_p2 <<<dim3(1, 32, 16), dim3(32),  0, stream>>>(Wq, KTV, P2);
  k_m  <<<dim3(8, 32, 2),  dim3(32),  0, stream>>>(P2, Wo, M);
  k_out<<<dim3(8, 32, 2),  dim3(256), 0, stream>>>(x, M, out);
}